// GINLayer_90031104459187
// MI455X (gfx1250) — compile-verified
//
#include <hip/hip_runtime.h>
#include <hip/hip_bf16.h>

#define D 128
#define BN_EPS 1e-5f
#define LDS_STRIDE 288  // floats per k-pair row: 1152B = 32 banks mod 64 ->
                        // even p on banks 0-31, odd p on banks 32-63

typedef __attribute__((ext_vector_type(2))) float v2f;
typedef __attribute__((ext_vector_type(8))) float v8f;

// ---------------------------------------------------------------------------
// h0 = (1 + eps) * x        (vectorized float4)
// ---------------------------------------------------------------------------
__global__ __launch_bounds__(256) void init_scale_kernel(
    const float4* __restrict__ x, const float* __restrict__ eps,
    float4* __restrict__ h0, int n4) {
  int i = blockIdx.x * blockDim.x + threadIdx.x;
  if (i >= n4) return;
  float s = 1.0f + eps[0];
  float4 v = x[i];
  h0[i] = make_float4(v.x * s, v.y * s, v.z * s, v.w * s);
}

// ---------------------------------------------------------------------------
// Scatter-add: one wave per edge, each lane owns 4 of the 128 features.
// h0[row] += vals[e] * x[col]   via native f32 global atomics.
// ---------------------------------------------------------------------------
__global__ __launch_bounds__(256) void edge_scatter_kernel(
    const float* __restrict__ x, const float* __restrict__ vals,
    const int* __restrict__ rows, const int* __restrict__ cols,
    float* __restrict__ h0, int E) {
  int t = blockIdx.x * blockDim.x + threadIdx.x;
  int e = t >> 5;          // one wave32 per edge
  int lane = t & 31;       // lane owns columns [4*lane, 4*lane+3]
  if (e >= E) return;
  int c = cols[e];
  int r = rows[e];
  float v = vals[e];
  const float4 xf = *(const float4*)(x + (size_t)c * D + lane * 4);
  float* dst = h0 + (size_t)r * D + lane * 4;
  unsafeAtomicAdd(dst + 0, v * xf.x);
  unsafeAtomicAdd(dst + 1, v * xf.y);
  unsafeAtomicAdd(dst + 2, v * xf.z);
  unsafeAtomicAdd(dst + 3, v * xf.w);
}

// ---------------------------------------------------------------------------
// Zero the 256-float stats buffer (sum[128], sumsq[128]).
// ---------------------------------------------------------------------------
__global__ void zero256_kernel(float* __restrict__ p) {
  p[threadIdx.x] = 0.0f;
}

// ---------------------------------------------------------------------------
// out[N,128] = A[N,128] @ W[128,128] + bias   using V_WMMA_F32_16X16X4_F32,
// with the per-column BN statistics (sum, sumsq) fused into the epilogue.
//
// Block = 256 threads = 8 waves covering 128 rows; each wave computes a
// 16-row stripe across all 128 columns (8 accumulator tiles).
//
// W is staged in LDS k-pair interleaved: Wlds[p*STRIDE + 2n + s] = W[2p+s][n]
// so a B fragment (lane&15 = col, half-wave selects k-pair p = 2kt+half) is a
// single aligned ds_load_b64 with no repacking movs, conflict-free by the
// 288-float row stride.
//
// A-frag (16x4 f32, ISA layout): lane&15 = M row; lanes<16 hold K={4kt,4kt+1},
// lanes>=16 hold K={4kt+2,4kt+3} -> one aligned float2 global load, software
// pipelined one k-step ahead.
// C/D (16x16 f32): VGPR r -> row r (+8 for upper half-wave), lane&15 -> col.
// ---------------------------------------------------------------------------
__global__ __launch_bounds__(256) void gemm_bias_stats_kernel(
    const float* __restrict__ A, const float* __restrict__ W,
    const float* __restrict__ bias, float* __restrict__ out,
    float* __restrict__ gstats, int N) {
  __shared__ float Wlds[64 * LDS_STRIDE];  // 72 KB

  const int tid = threadIdx.x;

  // Stage W (row-major in memory) into the k-pair interleaved LDS layout.
  // One wave covers exactly one k-row per iteration (32 float4 per row).
  {
    const float4* Wv = (const float4*)W;
#pragma unroll
    for (int it = 0; it < 16; ++it) {
      const int idx4 = tid + 256 * it;
      const float4 v = Wv[idx4];
      const int flat = idx4 * 4;
      const int k = flat >> 7;        // row of W
      const int n0 = flat & 127;      // first of 4 columns
      const int base = (k >> 1) * LDS_STRIDE + (k & 1);
      Wlds[base + (n0 + 0) * 2] = v.x;
      Wlds[base + (n0 + 1) * 2] = v.y;
      Wlds[base + (n0 + 2) * 2] = v.z;
      Wlds[base + (n0 + 3) * 2] = v.w;
    }
  }
  __syncthreads();

  const int wave = tid >> 5;
  const int lane = tid & 31;
  const int half = lane >> 4;  // which K-pair this half-wave supplies
  const int lm = lane & 15;
  const int mBase = blockIdx.x * 128 + wave * 16;

  // Clamp A row so out-of-range rows load valid (ignored) data: EXEC stays
  // all-ones through the WMMA loop as the ISA requires.
  int mrow = mBase + lm;
  if (mrow > N - 1) mrow = N - 1;
  const float* Arow = A + (size_t)mrow * D;

  // LDS base for this half-wave's B fragments: p = 2*kt + half.
  const float* Bbase = &Wlds[half * LDS_STRIDE + lm * 2];

  v8f acc[8] = {};

  v2f a_cur = *(const v2f*)(Arow + half * 2);  // kt = 0
  for (int kt = 0; kt < 32; ++kt) {
    // Prefetch next A fragment (clamped index: redundant load on last iter,
    // keeps the loop branch-free and EXEC uniform).
    const int ktn = (kt < 31) ? (kt + 1) : 31;
    const v2f a_next = *(const v2f*)(Arow + ktn * 4 + half * 2);
    const float* Bk = Bbase + (size_t)(2 * kt) * LDS_STRIDE;
#pragma unroll
    for (int nt = 0; nt < 8; ++nt) {
      const v2f b = *(const v2f*)(Bk + nt * 32);  // single ds_load_b64
      acc[nt] = __builtin_amdgcn_wmma_f32_16x16x4_f32(
          false, a_cur, false, b, (short)0, acc[nt], false, false);
    }
    a_cur = a_next;
  }

  // ---- Epilogue: bias add, store, fused per-column BN partial stats ----
  __syncthreads();               // done reading Wlds; reuse it for stats
  float* sstats = Wlds;          // [0..127]=sum, [128..255]=sumsq
  sstats[tid < 256 ? tid : 0] = 0.0f;
  __syncthreads();

#pragma unroll
  for (int nt = 0; nt < 8; ++nt) {
    const int n = nt * 16 + lm;
    const float bn = bias[n];
    float s = 0.0f, q = 0.0f;
#pragma unroll
    for (int r = 0; r < 8; ++r) {
      const int m = mBase + r + half * 8;
      const float v = acc[nt][r] + bn;
      if (m < N) {
        out[(size_t)m * D + n] = v;
        s += v;
        q += v * v;
      }
    }
    atomicAdd(&sstats[n], s);        // ds_add_f32 (LDS float atomic)
    atomicAdd(&sstats[D + n], q);
  }
  __syncthreads();

  // One global f32 atomic per column component per block.
  unsafeAtomicAdd(&gstats[tid], sstats[tid]);
}

// ---------------------------------------------------------------------------
// BatchNorm (biased var, training stats) + Swish, vectorized float4.
// mean = sum/N ; var = sumsq/N - mean^2 ; y = (v-mean)*rsqrt(var+eps)*g+be
// out = y * sigmoid(y)
// ---------------------------------------------------------------------------
__global__ __launch_bounds__(256) void bn_swish_kernel(
    const float4* __restrict__ t, const float* __restrict__ stats,
    const float* __restrict__ g, const float* __restrict__ be,
    float4* __restrict__ out, int n4, float invN) {
  int i = blockIdx.x * blockDim.x + threadIdx.x;
  if (i >= n4) return;
  const int c0 = (i & (D / 4 - 1)) * 4;  // starting column of this float4
  float4 v = t[i];
  float in[4] = {v.x, v.y, v.z, v.w};
  float o[4];
#pragma unroll
  for (int j = 0; j < 4; ++j) {
    const int c = c0 + j;
    const float mean = stats[c] * invN;
    const float var = stats[D + c] * invN - mean * mean;
    const float y = (in[j] - mean) * rsqrtf(var + BN_EPS) * g[c] + be[c];
    o[j] = y / (1.0f + __expf(-y));
  }
  out[i] = make_float4(o[0], o[1], o[2], o[3]);
}

// ---------------------------------------------------------------------------
// Launch: init -> scatter -> [zero stats, GEMM(+stats), BN+swish] x 2 layers
// ---------------------------------------------------------------------------
extern "C" void kernel_launch(void* const* d_in, const int* in_sizes, int n_in,
                              void* d_out, int out_size, void* d_ws, size_t ws_size,
                              hipStream_t stream) {
  const float* x    = (const float*)d_in[0];
  const float* vals = (const float*)d_in[1];
  const float* W1   = (const float*)d_in[2];
  const float* b1   = (const float*)d_in[3];
  const float* g1   = (const float*)d_in[4];
  const float* be1  = (const float*)d_in[5];
  const float* W2   = (const float*)d_in[6];
  const float* b2   = (const float*)d_in[7];
  const float* g2   = (const float*)d_in[8];
  const float* be2  = (const float*)d_in[9];
  const float* eps  = (const float*)d_in[10];
  const int* rows   = (const int*)d_in[11];
  const int* cols   = (const int*)d_in[12];
  float* out = (float*)d_out;

  const int N = in_sizes[0] / D;   // 50000
  const int E = in_sizes[1];       // 800000
  const int n4 = N * D / 4;
  const float invN = 1.0f / (float)N;

  float* buf0 = (float*)d_ws;                 // N*D
  float* buf1 = buf0 + (size_t)N * D;         // N*D
  float* stats = buf1 + (size_t)N * D;        // 256 floats

  const int elemBlocks = (n4 + 255) / 256;
  const int edgeBlocks = (E * 32 + 255) / 256;
  const int gemmBlocks = (N + 127) / 128;

  // h0 = (1+eps)*x, then scatter-add edge messages into it.
  init_scale_kernel<<<elemBlocks, 256, 0, stream>>>((const float4*)x, eps,
                                                    (float4*)buf0, n4);
  edge_scatter_kernel<<<edgeBlocks, 256, 0, stream>>>(x, vals, rows, cols,
                                                      buf0, E);

  // Layer 1: buf0 -> buf1 (GEMM+bias+stats) -> buf0 (BN+swish)
  zero256_kernel<<<1, 256, 0, stream>>>(stats);
  gemm_bias_stats_kernel<<<gemmBlocks, 256, 0, stream>>>(buf0, W1, b1, buf1,
                                                         stats, N);
  bn_swish_kernel<<<elemBlocks, 256, 0, stream>>>((const float4*)buf1, stats,
                                                  g1, be1, (float4*)buf0, n4,
                                                  invN);

  // Layer 2: buf0 -> buf1 (GEMM+bias+stats) -> d_out (BN+swish)
  zero256_kernel<<<1, 256, 0, stream>>>(stats);
  gemm_bias_stats_kernel<<<gemmBlocks, 256, 0, stream>>>(buf0, W2, b2, buf1,
                                                         stats, N);
  bn_swish_kernel<<<elemBlocks, 256, 0, stream>>>((const float4*)buf1, stats,
                                                  g2, be2, (float4*)out, n4,
                                                  invN);
}